// Generator_26723286515895
// MI455X (gfx1250) — compile-verified
//
#include <hip/hip_runtime.h>

// Affine grid_sample (bilinear, zero-padded) for B=128, C=3, H=W=256, fp32.
// Memory-bound gather: no dense GEMM structure -> WMMA/TDM inapplicable.
// Optimized for HBM/L2 bandwidth: wave32-coalesced b128 NT stores, shared
// gather indices across channels, v_med3 clamps, scalar per-batch params.

typedef float v4f __attribute__((ext_vector_type(4)));

constexpr int Bn = 128, Cn = 3, Hn = 256, Wn = 256;
constexpr int HW = Hn * Wn;

__global__ __launch_bounds__(256, 4) void affine_grid_sample_26723286515895(
    const float* __restrict__ img,
    const float* __restrict__ rot,
    const float* __restrict__ trans,
    const float* __restrict__ scale,
    const float* __restrict__ shear,
    float* __restrict__ out)
{
    const int b     = blockIdx.z;                       // uniform per block -> SGPR
    const int y     = (blockIdx.y << 2) + threadIdx.y;  // 0..255
    const int xbase = threadIdx.x << 2;                 // 4 pixels per thread

    // Per-batch affine parameters (uniform addresses -> scalar loads).
    const float r   = rot[b];
    const float s   = scale[b];
    const float tx  = trans[2 * b + 0];
    const float ty  = trans[2 * b + 1];
    const float shx = shear[2 * b + 0];
    const float shy = shear[2 * b + 1];

    const float cr = cosf(r);
    const float sr = sinf(r);
    const float a00 = s * (cr - sr * shy);
    const float a01 = s * (cr * shx - sr);
    const float a10 = s * (sr + cr * shy);
    const float a11 = s * (sr * shx + cr);

    // gy is constant along the row this thread covers.
    const float gy = 2.0f * ((float)y + 0.5f) * (1.0f / (float)Hn) - 1.0f;
    const float cx = fmaf(a01, gy, tx);   // grid_x = a00*gx + cx
    const float cy = fmaf(a11, gy, ty);   // grid_y = a10*gx + cy

    const float* p0 = img + (size_t)b * (Cn * HW);
    const float* p1 = p0 + HW;
    const float* p2 = p0 + 2 * HW;

    v4f r0, r1, r2;

#pragma unroll
    for (int p = 0; p < 4; ++p) {
        const int   x  = xbase + p;
        const float gx = 2.0f * ((float)x + 0.5f) * (1.0f / (float)Wn) - 1.0f;

        const float gridx = fmaf(a00, gx, cx);
        const float gridy = fmaf(a10, gx, cy);

        // Same form as reference: ((g+1)*W - 1) / 2
        const float ixf = ((gridx + 1.0f) * (float)Wn - 1.0f) * 0.5f;
        const float iyf = ((gridy + 1.0f) * (float)Hn - 1.0f) * 0.5f;

        const float x0f = floorf(ixf);
        const float y0f = floorf(iyf);
        const float x1f = x0f + 1.0f;
        const float y1f = y0f + 1.0f;

        const float wx1 = ixf - x0f, wx0 = 1.0f - wx1;
        const float wy1 = iyf - y0f, wy0 = 1.0f - wy1;

        // Separable validity masks (weights are >= 0, so masked product
        // equals reference's w * valid exactly).
        const float wx0m = (x0f >= 0.0f && x0f <= 255.0f) ? wx0 : 0.0f;
        const float wx1m = (x1f >= 0.0f && x1f <= 255.0f) ? wx1 : 0.0f;
        const float wy0m = (y0f >= 0.0f && y0f <= 255.0f) ? wy0 : 0.0f;
        const float wy1m = (y1f >= 0.0f && y1f <= 255.0f) ? wy1 : 0.0f;

        const float w00 = wx0m * wy0m;
        const float w10 = wx1m * wy0m;
        const float w01 = wx0m * wy1m;
        const float w11 = wx1m * wy1m;

        // Clamp-then-cast, single v_med3_f32 each.
        const int x0i = (int)__builtin_amdgcn_fmed3f(x0f, 0.0f, 255.0f);
        const int x1i = (int)__builtin_amdgcn_fmed3f(x1f, 0.0f, 255.0f);
        const int y0i = (int)__builtin_amdgcn_fmed3f(y0f, 0.0f, 255.0f);
        const int y1i = (int)__builtin_amdgcn_fmed3f(y1f, 0.0f, 255.0f);

        const int i00 = (y0i << 8) | x0i;
        const int i10 = (y0i << 8) | x1i;
        const int i01 = (y1i << 8) | x0i;
        const int i11 = (y1i << 8) | x1i;

        if (p == 0) {
            // Warm channel-1/2 planes' corner lines (global_prefetch_b8).
            __builtin_prefetch(p1 + i00, 0, 0);
            __builtin_prefetch(p2 + i00, 0, 0);
        }

        // Indices/weights shared by all 3 channels (plane stride = 64K elems).
        r0[p] = fmaf(p0[i11], w11, fmaf(p0[i01], w01, fmaf(p0[i10], w10, p0[i00] * w00)));
        r1[p] = fmaf(p1[i11], w11, fmaf(p1[i01], w01, fmaf(p1[i10], w10, p1[i00] * w00)));
        r2[p] = fmaf(p2[i11], w11, fmaf(p2[i01], w01, fmaf(p2[i10], w10, p2[i00] * w00)));
    }

    // Output is write-once -> non-temporal b128 stores (don't pollute L2,
    // which is holding the input image for neighbor-pixel reuse).
    float* obase = out + (((size_t)(b * Cn) * Hn + y) * Wn + xbase);
    __builtin_nontemporal_store(r0, (v4f*)(obase));
    __builtin_nontemporal_store(r1, (v4f*)(obase + HW));
    __builtin_nontemporal_store(r2, (v4f*)(obase + 2 * HW));
}

extern "C" void kernel_launch(void* const* d_in, const int* in_sizes, int n_in,
                              void* d_out, int out_size, void* d_ws, size_t ws_size,
                              hipStream_t stream) {
    (void)in_sizes; (void)n_in; (void)out_size; (void)d_ws; (void)ws_size;

    const float* img   = (const float*)d_in[0];  // [128,3,256,256]
    const float* rot   = (const float*)d_in[1];  // [128,1]
    const float* trans = (const float*)d_in[2];  // [128,2]
    const float* scale = (const float*)d_in[3];  // [128,1]
    const float* shear = (const float*)d_in[4];  // [128,2]
    float* outp = (float*)d_out;                 // [128,3,256,256]

    dim3 block(64, 4, 1);          // 256 threads = 8 wave32, one batch per block
    dim3 grid(1, Hn / 4, Bn);      // y-tiles x batch

    affine_grid_sample_26723286515895<<<grid, block, 0, stream>>>(
        img, rot, trans, scale, shear, outp);
}